// CNNStencilModel_65266323030466
// MI455X (gfx1250) — compile-verified
//
#include <hip/hip_runtime.h>

typedef __attribute__((ext_vector_type(2))) float v2f;
typedef __attribute__((ext_vector_type(8))) float v8f;
typedef __attribute__((ext_vector_type(2))) unsigned int v2u;

#define HIDDEN 15
#define NL 5                    // hidden 1x1 layers (DEPTH-1)
#define B_DIM 1024
#define X_DIM 8192
#define XO (X_DIM - 1)          // 8191 outputs per batch row
#define TILES_PER_ROW 512       // ceil(8191/16)
#define NWAVES 16384            // 2048 blocks * 8 waves
#define ROWS_PER_STEP 32        // NWAVES / TILES_PER_ROW
#define ITERS 32                // B_DIM*TILES_PER_ROW / NWAVES
#define FLT_BIG 3.402823466e38f

// lane L <-> lane L^16 broadcast-swap (single register), v_permlanex16_b32
static __device__ __forceinline__ float xor16(float v) {
    return __uint_as_float(__builtin_amdgcn_permlanex16(
        0u, __float_as_uint(v), 0x76543210u, 0xfedcba98u, true, false));
}

// dual-destination half swap: a' = {a.row0, b.row0}, b' = {a.row1, b.row1}
// (v_permlane16_swap_b32 -- one instruction replaces permlanex16+cndmask pair)
static __device__ __forceinline__ void swap16(float& a, float& b) {
    v2u r = __builtin_amdgcn_permlane16_swap(__float_as_uint(a), __float_as_uint(b),
                                             false, false);
    a = __uint_as_float(r.x);
    b = __uint_as_float(r.y);
}

// single-instruction ReLU / clip via v_med3_f32
static __device__ __forceinline__ float relu1(float x) {
    return __builtin_amdgcn_fmed3f(x, 0.0f, FLT_BIG);
}
static __device__ __forceinline__ float clip01(float x) {
    return __builtin_amdgcn_fmed3f(x, 0.0f, 1.0f);
}

__global__ __launch_bounds__(256) void cnn_stencil_wmma(
    const float* __restrict__ rho,
    const float* __restrict__ w0,
    const float* __restrict__ b0,
    const float* __restrict__ Wh,
    const float* __restrict__ bh,
    const float* __restrict__ Wl,
    const float* __restrict__ bl,
    float* __restrict__ out)
{
    const int lane = threadIdx.x & 31;
    const int n    = lane & 15;     // position within tile / M index for A
    const int h    = lane >> 4;     // lane half (selects K pair / channel half)
    const bool hi  = (h != 0);

    // ---------- per-wave weight preload (15 padded to 16 with zeros) ----------
    // A operand (16x4 f32 layout): lane m=lane&15 holds W[m][4j+2h], W[m][4j+2h+1]
    v2f  aW[NL][4];
    float bhl[NL][8];
    const int m = n;
    for (int i = 0; i < NL; ++i) {
        for (int j = 0; j < 4; ++j) {
            const int k0 = 4 * j + 2 * h;
            float x0v = 0.f, x1v = 0.f;
            if (m < HIDDEN) {
                if (k0     < HIDDEN) x0v = Wh[i * HIDDEN * HIDDEN + m * HIDDEN + k0];
                if (k0 + 1 < HIDDEN) x1v = Wh[i * HIDDEN * HIDDEN + m * HIDDEN + k0 + 1];
            }
            aW[i][j].x = x0v; aW[i][j].y = x1v;
        }
        for (int g = 0; g < 8; ++g) {
            const int ch = g + 8 * h;
            bhl[i][g] = (ch < HIDDEN) ? bh[i * HIDDEN + ch] : 0.f;
        }
    }
    // first conv (k=2) weights ordered for the B-operand channel c = 4j + 2h + t
    v2f w0a[4], w0b[4], b0v[4];
    for (int j = 0; j < 4; ++j) {
        const int c0 = 4 * j + 2 * h;
        float a0 = 0.f, a1 = 0.f, q0 = 0.f, q1 = 0.f, z0 = 0.f, z1 = 0.f;
        if (c0 < HIDDEN)     { a0 = w0[c0 * 2 + 0];       q0 = w0[c0 * 2 + 1];       z0 = b0[c0];     }
        if (c0 + 1 < HIDDEN) { a1 = w0[(c0 + 1) * 2 + 0]; q1 = w0[(c0 + 1) * 2 + 1]; z1 = b0[c0 + 1]; }
        w0a[j].x = a0; w0a[j].y = a1;
        w0b[j].x = q0; w0b[j].y = q1;
        b0v[j].x = z0; b0v[j].y = z1;
    }
    float wl[8];
    for (int g = 0; g < 8; ++g) {
        const int ch = g + 8 * h;
        wl[g] = (ch < HIDDEN) ? Wl[ch] : 0.f;
    }
    const float blv = bl[0];

    // ---------- strided tile walk: everything but the offsets is invariant ----------
    const int waveId = blockIdx.x * (blockDim.x >> 5) + (threadIdx.x >> 5); // 0..NWAVES-1
    const int brow   = waveId >> 9;                 // starting batch row
    const int tx     = waveId & (TILES_PER_ROW - 1);
    const int x      = (tx << 4) + n;
    const int xc     = (x + 1 < X_DIM) ? (x + 1) : x;   // clamped neighbor (dead column)
    const bool doStore = (!hi) && (x < XO);

    // 32-bit element offsets (max ~8.4M elements, fits easily) -> v_add_nc_u32 bumps
    int o0 = brow * X_DIM + x;
    int o1 = brow * X_DIM + xc;
    int oo = brow * XO + x;
    const int RSTRIDE = ROWS_PER_STEP * X_DIM;
    const int OSTRIDE = ROWS_PER_STEP * XO;

    // software pipeline: current r0/r1 in flight while computing previous tile
    float r0 = rho[o0];
    float r1 = rho[o1];

    for (int it = 0; it < ITERS; ++it) {
        o0 += RSTRIDE;
        o1 += RSTRIDE;
        float nr0 = 0.f, nr1 = 0.f;
        if (it != ITERS - 1) {          // uniform branch; last iter would be OOB
            nr0 = rho[o0];
            nr1 = rho[o1];
        }

        // first conv (kernel=2) computed directly into B-operand layout
        const v2f r0v = {r0, r0};
        const v2f r1v = {r1, r1};
        v2f Bm[4];
        for (int j = 0; j < 4; ++j) {
            v2f t = w0a[j] * r0v + b0v[j];   // pk_fma
            t     = w0b[j] * r1v + t;        // pk_fma
            Bm[j].x = relu1(t.x);
            Bm[j].y = relu1(t.y);
        }

        float d[8];
        for (int i = 0; i < NL; ++i) {
            v8f c;
            for (int g = 0; g < 8; ++g) c[g] = bhl[i][g];   // bias per out-channel row
            c = __builtin_amdgcn_wmma_f32_16x16x4_f32(false, aW[i][0], false, Bm[0], (short)0, c, false, false);
            c = __builtin_amdgcn_wmma_f32_16x16x4_f32(false, aW[i][1], false, Bm[1], (short)0, c, false, false);
            c = __builtin_amdgcn_wmma_f32_16x16x4_f32(false, aW[i][2], false, Bm[2], (short)0, c, false, false);
            c = __builtin_amdgcn_wmma_f32_16x16x4_f32(false, aW[i][3], false, Bm[3], (short)0, c, false, false);
            for (int g = 0; g < 8; ++g) d[g] = relu1(c[g]);

            if (i + 1 < NL) {
                // D layout -> next-layer B operand: 4 dual-destination half swaps
                swap16(d[0], d[2]);
                swap16(d[1], d[3]);
                swap16(d[4], d[6]);
                swap16(d[5], d[7]);
                Bm[0].x = d[0]; Bm[0].y = d[1];
                Bm[2].x = d[2]; Bm[2].y = d[3];
                Bm[1].x = d[4]; Bm[1].y = d[5];
                Bm[3].x = d[6]; Bm[3].y = d[7];
            }
        }

        // final 15->1 projection + clip
        float partial = 0.f;
        for (int g = 0; g < 8; ++g) partial += d[g] * wl[g];
        float tot = partial + xor16(partial) + blv;
        tot = clip01(tot);
        if (doStore) out[oo] = tot;
        oo += OSTRIDE;

        r0 = nr0;
        r1 = nr1;
    }
}

extern "C" void kernel_launch(void* const* d_in, const int* in_sizes, int n_in,
                              void* d_out, int out_size, void* d_ws, size_t ws_size,
                              hipStream_t stream) {
    const float* rho = (const float*)d_in[0];
    const float* w0  = (const float*)d_in[1];
    const float* b0  = (const float*)d_in[2];
    const float* Wh  = (const float*)d_in[3];
    const float* bh  = (const float*)d_in[4];
    const float* Wl  = (const float*)d_in[5];
    const float* bl  = (const float*)d_in[6];
    float* out = (float*)d_out;

    dim3 grid(2048);   // 16384 waves -> exactly 32 tiles per wave (constant trip count)
    dim3 block(256);   // 8 waves per workgroup
    hipLaunchKernelGGL(cnn_stencil_wmma, grid, block, 0, stream,
                       rho, w0, b0, Wh, bh, Wl, bl, out);
}